// SAGPooling_Global_76089640616130
// MI455X (gfx1250) — compile-verified
//
#include <hip/hip_runtime.h>
#include <math.h>

// ---------------- problem constants (from reference) ----------------
constexpr int NUM_GRAPHS = 64;
constexpr int NPG        = 1024;            // nodes per graph
constexpr int NT         = NUM_GRAPHS * NPG; // 65536 total nodes
constexpr int HID        = 128;
constexpr int KPOOL      = 512;

typedef float v2f __attribute__((ext_vector_type(2)));
typedef float v8f __attribute__((ext_vector_type(8)));

// Only check on the DEVICE pass — host pass never has AMDGCN builtins.
#if defined(__HIP_DEVICE_COMPILE__) && !__has_builtin(__builtin_amdgcn_wmma_f32_16x16x4_f32)
#error "device pass: missing __builtin_amdgcn_wmma_f32_16x16x4_f32"
#endif

__device__ __forceinline__ void atomicAddF(float* p, float v) {
  __hip_atomic_fetch_add(p, v, __ATOMIC_RELAXED, __HIP_MEMORY_SCOPE_AGENT);
}

// ---------------- degree kernels ----------------
__global__ void k_zero(float* p, int n) {
  int i = blockIdx.x * blockDim.x + threadIdx.x;
  if (i < n) p[i] = 0.0f;
}

__global__ void k_count_deg(const int* __restrict__ ei, float* __restrict__ degf, int E) {
  int e = blockIdx.x * blockDim.x + threadIdx.x;
  if (e < E) atomicAddF(&degf[ei[E + e]], 1.0f);  // dst = ei[E+e]
}

__global__ void k_deg_fin(const float* __restrict__ degf, float* __restrict__ dinv,
                          float* __restrict__ invd, int n) {
  int i = blockIdx.x * blockDim.x + threadIdx.x;
  if (i < n) {
    float d = degf[i] + 1.0f;     // +1 self loop
    dinv[i] = rsqrtf(d);
    invd[i] = 1.0f / d;
  }
}

// ---------------- WMMA fp32 GEMM: H[nr x 128] = X[nr x 128] @ W[128 x 128] ----------------
// block = 256 threads (8 waves); block covers 128 rows; wave = 16 rows x 128 cols.
__global__ __launch_bounds__(256)
void k_gemm128(const float* __restrict__ X, const float* __restrict__ W,
               float* __restrict__ H) {
  // LDS-staged W, pre-swizzled into B-fragment order:
  // slot = ((k*8 + n)*32 + lane), holds float2 = W[4k+2*(lane>>4)+{0,1}][16n + (lane&15)]
  __shared__ float sW[128 * 128];
  const int t = threadIdx.x;
  for (int idx = t; idx < 128 * 128 / 2; idx += 256) {
    int lane = idx & 31;
    int kn   = idx >> 5;
    int n    = kn & 7;
    int k    = kn >> 3;
    int col  = n * 16 + (lane & 15);
    int kk   = k * 4 + 2 * (lane >> 4);
    sW[idx * 2 + 0] = W[kk * 128 + col];
    sW[idx * 2 + 1] = W[(kk + 1) * 128 + col];
  }
  __syncthreads();

  const int wave = t >> 5;
  const int lane = t & 31;
  const int m    = lane & 15;     // row within tile / col within N-tile
  const int kh   = lane >> 4;     // lane-half selects K pair
  const int row0 = blockIdx.x * 128 + wave * 16;

  // Preload A fragments: af[k] = X[row0+m][4k + 2*kh + {0,1}]
  const float* xrow = X + (size_t)(row0 + m) * 128 + 2 * kh;
  v2f af[32];
#pragma unroll
  for (int k = 0; k < 32; ++k) af[k] = *(const v2f*)(xrow + 4 * k);

  v8f zero = {};
  v8f acc[8];
#pragma unroll
  for (int n = 0; n < 8; ++n) acc[n] = zero;

#if defined(__HIP_DEVICE_COMPILE__)
#pragma unroll
  for (int k = 0; k < 32; ++k) {
#pragma unroll
    for (int n = 0; n < 8; ++n) {
      v2f b = *(const v2f*)&sW[((k * 8 + n) * 32 + lane) * 2];
      acc[n] = __builtin_amdgcn_wmma_f32_16x16x4_f32(
          false, af[k], false, b, (short)0, acc[n], false, false);
    }
  }
#endif

  // C/D layout: vgpr v, lanes 0-15 -> M=v, lanes 16-31 -> M=v+8 ; N = lane&15
  const int rbase = row0 + 8 * kh;
#pragma unroll
  for (int n = 0; n < 8; ++n)
#pragma unroll
    for (int v = 0; v < 8; ++v)
      H[(size_t)(rbase + v) * 128 + n * 16 + m] = acc[n][v];
}

// ---------------- conv epilogue pieces ----------------
// out = H * invdeg[node] + bias[f]  (self-loop term + bias)
__global__ void k_init_agg(const float* __restrict__ H, const float* __restrict__ invd,
                           const float* __restrict__ bias, float* __restrict__ out, int n) {
  int i = blockIdx.x * blockDim.x + threadIdx.x;
  if (i < n) out[i] = H[i] * invd[i >> 7] + bias[i & 127];
}

// agg[dst] += H[src] * dinv[src]*dinv[dst]   (one wave per edge, float4 per lane)
__global__ __launch_bounds__(256)
void k_edge_scatter(const float* __restrict__ H, const int* __restrict__ ei,
                    const float* __restrict__ dinv, float* __restrict__ agg, int E) {
  int e = blockIdx.x * 8 + (threadIdx.x >> 5);
  if (e >= E) return;
  int lane = threadIdx.x & 31;
  int s = ei[e], d = ei[E + e];
  float w = dinv[s] * dinv[d];
  float4 hv = ((const float4*)(H + (size_t)s * 128))[lane];
  float* ad = agg + (size_t)d * 128 + lane * 4;
  atomicAddF(ad + 0, hv.x * w);
  atomicAddF(ad + 1, hv.y * w);
  atomicAddF(ad + 2, hv.z * w);
  atomicAddF(ad + 3, hv.w * w);
}

__global__ void k_relu(float* p, int n) {
  int i = blockIdx.x * blockDim.x + threadIdx.x;
  if (i < n) p[i] = fmaxf(p[i], 0.0f);
}

// ---------------- score conv (384 -> 1) ----------------
__global__ __launch_bounds__(256)
void k_score_gemm(const float* __restrict__ x1, const float* __restrict__ x2,
                  const float* __restrict__ x3, const float* __restrict__ Ws,
                  float* __restrict__ hs) {
  int node = blockIdx.x * 8 + (threadIdx.x >> 5);
  int lane = threadIdx.x & 31;
  size_t b = (size_t)node * 128;
  float s = 0.0f;
#pragma unroll
  for (int j = lane; j < 128; j += 32)
    s += x1[b + j] * Ws[j] + x2[b + j] * Ws[128 + j] + x3[b + j] * Ws[256 + j];
#pragma unroll
  for (int off = 16; off > 0; off >>= 1) s += __shfl_down(s, off, 32);
  if (lane == 0) hs[node] = s;
}

__global__ void k_score_init(const float* __restrict__ hs, const float* __restrict__ invd,
                             const float* __restrict__ bs, float* __restrict__ aggs, int n) {
  int i = blockIdx.x * blockDim.x + threadIdx.x;
  if (i < n) aggs[i] = hs[i] * invd[i] + bs[0];
}

__global__ void k_score_edge(const float* __restrict__ hs, const int* __restrict__ ei,
                             const float* __restrict__ dinv, float* __restrict__ aggs, int E) {
  int e = blockIdx.x * blockDim.x + threadIdx.x;
  if (e < E) {
    int s = ei[e], d = ei[E + e];
    atomicAddF(&aggs[d], hs[s] * dinv[s] * dinv[d]);
  }
}

__global__ void k_tanh(float* p, int n) {
  int i = blockIdx.x * blockDim.x + threadIdx.x;
  if (i < n) p[i] = tanhf(p[i]);
}

// ---------------- per-graph top-k by rank counting ----------------
__global__ __launch_bounds__(1024)
void k_topk(const float* __restrict__ score, int* __restrict__ seli,
            float* __restrict__ selv) {
  __shared__ float ss[NPG];
  int g = blockIdx.x, i = threadIdx.x;
  ss[i] = score[g * NPG + i];
  __syncthreads();
  float mine = ss[i];
  int rank = 0;
  for (int j = 0; j < NPG; ++j) {
    float o = ss[j];
    rank += (o > mine) || (o == mine && j < i);   // lax.top_k tie-break: lower index first
  }
  if (rank < KPOOL) {
    seli[g * KPOOL + rank] = g * NPG + i;
    selv[g * KPOOL + rank] = mine;
  }
}

// ---------------- pooled features: [max | mean] over selected nodes ----------------
__global__ __launch_bounds__(384)
void k_pool(const float* __restrict__ x1, const float* __restrict__ x2,
            const float* __restrict__ x3, const int* __restrict__ seli,
            const float* __restrict__ selv, float* __restrict__ gf) {
  int g = blockIdx.x, f = threadIdx.x;
  const float* src = (f < 128) ? x1 : (f < 256) ? x2 : x3;
  int fo = (f < 128) ? f : (f < 256) ? f - 128 : f - 256;
  float mx = -INFINITY, sm = 0.0f;
  for (int r = 0; r < KPOOL; ++r) {
    int node = seli[g * KPOOL + r];
    float v  = src[(size_t)node * 128 + fo] * selv[g * KPOOL + r];
    mx = fmaxf(mx, v);
    sm += v;
  }
  gf[g * 768 + f]       = mx;
  gf[g * 768 + 384 + f] = sm * (1.0f / KPOOL);
}

// ---------------- MLP head + log_softmax, one block per graph ----------------
__global__ __launch_bounds__(128)
void k_head(const float* __restrict__ gf,
            const float* __restrict__ Wl1, const float* __restrict__ bl1,
            const float* __restrict__ Wl2, const float* __restrict__ bl2,
            const float* __restrict__ Wc,  const float* __restrict__ bc,
            float* __restrict__ out) {
  __shared__ float sg[768];
  __shared__ float sh1[128];
  __shared__ float sh2[64];
  __shared__ float sl[10];
  int g = blockIdx.x, t = threadIdx.x;
  for (int j = t; j < 768; j += 128) sg[j] = gf[g * 768 + j];
  __syncthreads();
  {
    float a = bl1[t];
    for (int k = 0; k < 768; ++k) a += sg[k] * Wl1[k * 128 + t];
    sh1[t] = fmaxf(a, 0.0f);
  }
  __syncthreads();
  if (t < 64) {
    float a = bl2[t];
    for (int k = 0; k < 128; ++k) a += sh1[k] * Wl2[k * 64 + t];
    sh2[t] = fmaxf(a, 0.0f);
  }
  __syncthreads();
  if (t < 10) {
    float a = bc[t];
    for (int k = 0; k < 64; ++k) a += sh2[k] * Wc[k * 10 + t];
    sl[t] = a;
  }
  __syncthreads();
  if (t == 0) {
    float m = sl[0];
    for (int c = 1; c < 10; ++c) m = fmaxf(m, sl[c]);
    float se = 0.0f;
    for (int c = 0; c < 10; ++c) se += expf(sl[c] - m);
    float lse = m + logf(se);
    for (int c = 0; c < 10; ++c) out[g * 10 + c] = sl[c] - lse;
  }
}

// ---------------- launcher ----------------
extern "C" void kernel_launch(void* const* d_in, const int* in_sizes, int n_in,
                              void* d_out, int out_size, void* d_ws, size_t ws_size,
                              hipStream_t stream) {
  const float* x   = (const float*)d_in[0];
  const int*   ei  = (const int*)d_in[1];     // [2,E]: src then dst
  const float* W1  = (const float*)d_in[3];
  const float* b1  = (const float*)d_in[4];
  const float* W2  = (const float*)d_in[5];
  const float* b2  = (const float*)d_in[6];
  const float* W3  = (const float*)d_in[7];
  const float* b3  = (const float*)d_in[8];
  const float* Ws  = (const float*)d_in[9];
  const float* bs  = (const float*)d_in[10];
  const float* Wl1 = (const float*)d_in[11];
  const float* bl1 = (const float*)d_in[12];
  const float* Wl2 = (const float*)d_in[13];
  const float* bl2 = (const float*)d_in[14];
  const float* Wc  = (const float*)d_in[15];
  const float* bc  = (const float*)d_in[16];
  float* out = (float*)d_out;

  const int E = in_sizes[1] / 2;

  // workspace carve
  float* p    = (float*)d_ws;
  float* H    = p; p += (size_t)NT * 128;
  float* x1   = p; p += (size_t)NT * 128;
  float* x2   = p; p += (size_t)NT * 128;
  float* x3   = p; p += (size_t)NT * 128;
  float* degf = p; p += NT;
  float* dinv = p; p += NT;
  float* invd = p; p += NT;
  float* hs   = p; p += NT;
  float* aggs = p; p += NT;
  float* selv = p; p += NUM_GRAPHS * KPOOL;
  float* gf   = p; p += NUM_GRAPHS * 768;
  int*   seli = (int*)p;

  const int NF = NT * 128;

  // degree / norms (shared by all convs)
  k_zero<<<(NT + 255) / 256, 256, 0, stream>>>(degf, NT);
  k_count_deg<<<(E + 255) / 256, 256, 0, stream>>>(ei, degf, E);
  k_deg_fin<<<(NT + 255) / 256, 256, 0, stream>>>(degf, dinv, invd, NT);

  // conv1: x -> x1
  k_gemm128<<<NT / 128, 256, 0, stream>>>(x, W1, H);
  k_init_agg<<<(NF + 255) / 256, 256, 0, stream>>>(H, invd, b1, x1, NF);
  k_edge_scatter<<<(E + 7) / 8, 256, 0, stream>>>(H, ei, dinv, x1, E);
  k_relu<<<(NF + 255) / 256, 256, 0, stream>>>(x1, NF);

  // conv2: x1 -> x2
  k_gemm128<<<NT / 128, 256, 0, stream>>>(x1, W2, H);
  k_init_agg<<<(NF + 255) / 256, 256, 0, stream>>>(H, invd, b2, x2, NF);
  k_edge_scatter<<<(E + 7) / 8, 256, 0, stream>>>(H, ei, dinv, x2, E);
  k_relu<<<(NF + 255) / 256, 256, 0, stream>>>(x2, NF);

  // conv3: x2 -> x3
  k_gemm128<<<NT / 128, 256, 0, stream>>>(x2, W3, H);
  k_init_agg<<<(NF + 255) / 256, 256, 0, stream>>>(H, invd, b3, x3, NF);
  k_edge_scatter<<<(E + 7) / 8, 256, 0, stream>>>(H, ei, dinv, x3, E);
  k_relu<<<(NF + 255) / 256, 256, 0, stream>>>(x3, NF);

  // score conv + tanh
  k_score_gemm<<<NT / 8, 256, 0, stream>>>(x1, x2, x3, Ws, hs);
  k_score_init<<<(NT + 255) / 256, 256, 0, stream>>>(hs, invd, bs, aggs, NT);
  k_score_edge<<<(E + 255) / 256, 256, 0, stream>>>(hs, ei, dinv, aggs, E);
  k_tanh<<<(NT + 255) / 256, 256, 0, stream>>>(aggs, NT);

  // top-k, pooling, head
  k_topk<<<NUM_GRAPHS, 1024, 0, stream>>>(aggs, seli, selv);
  k_pool<<<NUM_GRAPHS, 384, 0, stream>>>(x1, x2, x3, seli, selv, gf);
  k_head<<<NUM_GRAPHS, 128, 0, stream>>>(gf, Wl1, bl1, Wl2, bl2, Wc, bc, out);
  (void)n_in; (void)out_size; (void)ws_size;
}